// RowTemporalWeight_51994874086000
// MI455X (gfx1250) — compile-verified
//
#include <hip/hip_runtime.h>
#include <math.h>

// Shapes are fixed by the reference: x [16, 64, 256, 256] f32.
#define B_   16
#define C_   64
#define H_   256
#define W_   256
#define HW_  (H_ * W_)          // 65536 = 2^16
#define CHW_ (C_ * HW_)

typedef float v2f __attribute__((ext_vector_type(2)));
typedef float v8f __attribute__((ext_vector_type(8)));

// ---------------------------------------------------------------------------
// Stage 1: per-position channel L2 norm via V_WMMA_F32_16X16X4_F32.
// Each wave handles 16 consecutive flattened positions p = b*HW + h*W + w
// (tiles never straddle a batch boundary since HW % 16 == 0).
// A-matrix 16x4 f32 layout: lane L -> M = L&15; VGPR0 = A[M, 2*(L>>4)],
// VGPR1 = A[M, 2*(L>>4)+1]. B = all-ones (layout-independent), so
// D[m, n] = sum_k A[m, k] + C[m, n]: chained over 16 chunks of 4 channels
// this accumulates sum_c x^2 replicated across all 16 columns.
// C/D layout: acc[j] at lane L = D[j + 8*(L>>4), L&15].
// ---------------------------------------------------------------------------
__global__ __launch_bounds__(256) void rtw_norm_wmma(const float* __restrict__ x,
                                                     float* __restrict__ nrm) {
  const int lane  = threadIdx.x & 31;
  const int wave  = blockIdx.x * 8 + (threadIdx.x >> 5);
  const int p0    = wave << 4;                // first of 16 positions
  const int m     = lane & 15;                // row of A / position in tile
  const int khalf = lane >> 4;                // K half select
  const int pos   = p0 + m;
  const int b     = pos >> 16;                // / HW_
  const int q     = pos & (HW_ - 1);          // h*W + w
  const float* xb = x + b * CHW_ + q;         // lane-contiguous in q

  v8f acc = {0.f, 0.f, 0.f, 0.f, 0.f, 0.f, 0.f, 0.f};
  v2f Bones; Bones[0] = 1.0f; Bones[1] = 1.0f;

#pragma unroll
  for (int c0 = 0; c0 < C_; c0 += 4) {
    const int ca = c0 + (khalf << 1);
    const float a0 = xb[ca * HW_];
    const float a1 = xb[(ca + 1) * HW_];
    v2f A; A[0] = a0 * a0; A[1] = a1 * a1;
    // 8 args: (neg_a, A, neg_b, B, c_mod, C, reuse_a, reuse_b)
    acc = __builtin_amdgcn_wmma_f32_16x16x4_f32(false, A, false, Bones,
                                                (short)0, acc, false, false);
  }

  // Lane with (lane&15)==j holds row j (lanes<16) or row j+8 (lanes>=16)
  // in acc[j]; every column is identical, so write column j.
#pragma unroll
  for (int j = 0; j < 8; ++j) {
    if ((lane & 15) == j) {
      nrm[p0 + j + (khalf << 3)] = sqrtf(acc[j]);
    }
  }
}

// ---------------------------------------------------------------------------
// Stage 2: softmax over H per (b, w) column, in place (nrm -> weights).
// Block = (32 w-lanes, 8 h-slices). Loads are lane-contiguous in w.
// ---------------------------------------------------------------------------
__global__ __launch_bounds__(256) void rtw_softmax_h(float* __restrict__ nw) {
  const int tx = threadIdx.x;                 // w lane (0..31)
  const int ty = threadIdx.y;                 // h slice (0..7)
  const int b  = blockIdx.x >> 3;             // W_/32 = 8 chunks per batch
  const int w0 = (blockIdx.x & 7) << 5;
  float* col = nw + b * HW_ + w0 + tx;        // stride W_ along h

  __shared__ float redM[8][32];
  __shared__ float redS[8][32];

  float mx = -INFINITY;
  for (int h = ty; h < H_; h += 8)
    mx = fmaxf(mx, col[h * W_]);
  redM[ty][tx] = mx;
  __syncthreads();
  mx = redM[0][tx];
#pragma unroll
  for (int j = 1; j < 8; ++j) mx = fmaxf(mx, redM[j][tx]);

  float s = 0.0f;
  for (int h = ty; h < H_; h += 8)
    s += expf(col[h * W_] - mx);
  redS[ty][tx] = s;
  __syncthreads();
  s = 0.0f;
#pragma unroll
  for (int j = 0; j < 8; ++j) s += redS[j][tx];

  const float inv = 1.0f / s;
  for (int h = ty; h < H_; h += 8) {
    const int off = h * W_;
    col[off] = expf(col[off] - mx) * inv;     // softmax weight
  }
}

// ---------------------------------------------------------------------------
// Stage 3: per (b, c, w): weighted sum over H, divide by (weight_sum + 1e-8)
// exactly as the reference does, then broadcast-write across all H rows.
// Weights (4 MB) are L2-resident across the 64 c reuses.
// ---------------------------------------------------------------------------
__global__ __launch_bounds__(256) void rtw_compress_bcast(const float* __restrict__ x,
                                                          const float* __restrict__ wgt,
                                                          float* __restrict__ out) {
  const int idx = blockIdx.x * 256 + threadIdx.x;   // [0, B*C*W)
  const int w = idx & (W_ - 1);
  const int t = idx >> 8;
  const int c = t & (C_ - 1);
  const int b = t >> 6;

  const float* xp = x + b * CHW_ + c * HW_ + w;     // stride W_ in h
  const float* gp = wgt + b * HW_ + w;

  float acc = 0.0f, ws = 0.0f;
#pragma unroll 4
  for (int h = 0; h < H_; ++h) {
    const int off = h * W_;
    const float g = gp[off];
    acc = fmaf(xp[off], g, acc);
    ws += g;
  }
  const float comp = acc / (ws + 1e-8f);

  float* op = out + b * CHW_ + c * HW_ + w;
#pragma unroll 4
  for (int h = 0; h < H_; ++h) op[h * W_] = comp;
}

// ---------------------------------------------------------------------------
extern "C" void kernel_launch(void* const* d_in, const int* in_sizes, int n_in,
                              void* d_out, int out_size, void* d_ws, size_t ws_size,
                              hipStream_t stream) {
  (void)in_sizes; (void)n_in; (void)out_size; (void)ws_size;
  const float* x  = (const float*)d_in[0];
  float* out      = (float*)d_out;
  float* nw       = (float*)d_ws;   // B*H*W floats = 4 MB scratch

  // Stage 1: 65536 tiles of 16 positions, 8 waves per block.
  rtw_norm_wmma<<<(B_ * HW_ / 16) / 8, 256, 0, stream>>>(x, nw);
  // Stage 2: one block per (b, 32-wide w chunk).
  rtw_softmax_h<<<B_ * (W_ / 32), dim3(32, 8, 1), 0, stream>>>(nw);
  // Stage 3: one thread per (b, c, w).
  rtw_compress_bcast<<<(B_ * C_ * W_) / 256, 256, 0, stream>>>(x, nw, out);
}